// IdentityConvolution_72258529788710
// MI455X (gfx1250) — compile-verified
//
#include <hip/hip_runtime.h>

// Reference: r[b,p] = sum_c x_real[b,c,p]; i[b,p] = sum_c x_imag[b,c,p];
// out[b,cout,p] = complex(r, i) for all 64 cout. Memory-bound:
// 256MB read + 256MB write -> ~22us at 23.3 TB/s.
//
// Channel reduction done with V_WMMA_F32_16X16X4_F32, A = ones:
// D[m,n] = sum_k B[k,n] (+C). 16 chained WMMAs cover K=64 channels at full
// fp32 precision. With A = all-ones the result is invariant to the exact
// K<->(vgpr,half-wave) assignment of the B layout; only N = lane&15 matters.

typedef __attribute__((ext_vector_type(2))) float v2f;
typedef __attribute__((ext_vector_type(8))) float v8f;

#define HW   65536   // 256*256 pixels per (batch, channel) plane
#define CIN  64
#define COUT 64
#define TILES_PER_WAVE 8  // 32768 tiles / 4096 waves

__global__ __launch_bounds__(256) void
IdentityConvolution_gfx1250_kernel(const float* __restrict__ xr,
                                   const float* __restrict__ xi,
                                   float* __restrict__ out)
{
    const int lane = threadIdx.x & 31;
    const int wave = (blockIdx.x * (blockDim.x >> 5)) + (threadIdx.x >> 5);

    const int n = lane & 15;   // pixel column within 16-pixel tile (N of WMMA)
    const int h = lane >> 4;   // half-wave: selects channel pair / output parity

    const v2f ones = {1.0f, 1.0f};  // A-matrix (16x4 of 1.0) striped per lane

    for (int t = 0; t < TILES_PER_WAVE; ++t) {
        const int tile = wave * TILES_PER_WAVE + t;   // 0..32767
        const int b    = tile >> 12;                  // 4096 tiles per batch
        const int p0   = (tile & 4095) << 4;          // pixel base in batch

        // Per-lane base for this lane's pixel column and half-wave channels.
        const size_t ibase = (size_t)b * (CIN * HW) + (size_t)(2 * h) * HW
                             + (size_t)(p0 + n);
        const float* brh = xr + ibase;
        const float* bih = xi + ibase;

        v8f accr = {};
        v8f acci = {};
#pragma unroll
        for (int cg = 0; cg < 16; ++cg) {   // 4 channels per WMMA, K=64 total
            v2f bvr, bvi;
            bvr.x = brh[(size_t)(4 * cg + 0) * HW];
            bvr.y = brh[(size_t)(4 * cg + 1) * HW];
            bvi.x = bih[(size_t)(4 * cg + 0) * HW];
            bvi.y = bih[(size_t)(4 * cg + 1) * HW];
            // D = ones(16x4) x B(4x16) + C  => column sums accumulate in C
            accr = __builtin_amdgcn_wmma_f32_16x16x4_f32(
                false, ones, false, bvr, (short)0, accr, false, false);
            acci = __builtin_amdgcn_wmma_f32_16x16x4_f32(
                false, ones, false, bvi, (short)0, acci, false, false);
        }

        // Every row of D is identical; every lane holds the channel-sum of
        // pixel p0 + (lane&15) in acc[0]. Pack complex pair and broadcast to
        // 64 output channels: half-waves cover even/odd channels in parallel,
        // so each b64 store writes two full 128B channel segments.
        v2f val;
        val.x = accr[0];
        val.y = acci[0];

        v2f* ob = (v2f*)out + (size_t)b * (COUT * HW) + (size_t)h * HW
                  + (size_t)(p0 + n);
#pragma unroll
        for (int cg = 0; cg < 32; ++cg) {
            // streaming output: non-temporal, keep out of WGP$/L2
            __builtin_nontemporal_store(val, ob + (size_t)(2 * cg) * HW);
        }
    }
}

extern "C" void kernel_launch(void* const* d_in, const int* in_sizes, int n_in,
                              void* d_out, int out_size, void* d_ws, size_t ws_size,
                              hipStream_t stream) {
    (void)in_sizes; (void)n_in; (void)out_size; (void)d_ws; (void)ws_size;
    const float* xr = (const float*)d_in[0];
    const float* xi = (const float*)d_in[1];
    float* out = (float*)d_out;

    // 32768 tiles of 16 pixels; 8 tiles per wave -> 4096 waves ->
    // 512 blocks of 256 threads (8 wave32 per block, full EXEC for WMMA).
    IdentityConvolution_gfx1250_kernel<<<512, 256, 0, stream>>>(xr, xi, out);
}